// ProposalLayer_58823872086112
// MI455X (gfx1250) — compile-verified
//
#include <hip/hip_runtime.h>
#include <hip/hip_bf16.h>

// ---------------- problem constants (match reference) ----------------
#define BATCH 8
#define NA    261888
#define KSEL  6000          // PRE_NMS_LIMIT (== min(6000, NA))
#define NOUT  2000          // PROPOSAL_COUNT
#define NW    188           // ceil(KSEL/32) mask words per row
#define ECAP  8192          // capacity of tie-resolution list
#define NMS_T 0.7f

static_assert(NW * 32 >= KSEL, "mask width");
static_assert((KSEL % 16) == 0, "row tiles");
static_assert((KSEL % 4) == 0, "b128 chunks");

typedef __attribute__((ext_vector_type(2))) float v2f;
typedef __attribute__((ext_vector_type(8))) float v8f;

// ---------------- utility: zero a u32 region ----------------
__global__ void zero_u32_kernel(unsigned int* __restrict__ p, int n) {
  for (int i = blockIdx.x * blockDim.x + threadIdx.x; i < n;
       i += gridDim.x * blockDim.x)
    p[i] = 0u;
}

// ---------------- pass 1: histogram of high 16 score bits ----------------
// scores are softmax outputs in [0,1] -> nonnegative floats, so the raw bit
// pattern is monotone and we can radix-select directly on it.
__global__ void hist_hi_kernel(const float* __restrict__ probs,
                               unsigned int* __restrict__ hist) {
  const int n = BATCH * NA;
  for (int i = blockIdx.x * blockDim.x + threadIdx.x; i < n;
       i += gridDim.x * blockDim.x) {
    __builtin_prefetch(probs + 2 * i + 4096, 0, 0);   // global_prefetch_b8
    const unsigned int bits = __float_as_uint(probs[2 * i + 1]);
    const int b = i / NA;
    atomicAdd(&hist[(b << 16) + (bits >> 16)], 1u);
  }
}

// params layout per batch (8 u32): [0]=hiBin [1]=cntAbovePrefix [2]=needInBin
// [3]=vstar bits [4]=cntStrictlyGreater [5]=numEqualsNeeded [6]=candCtr [7]=eqCtr
__global__ void find_hi_kernel(const unsigned int* __restrict__ hist,
                               unsigned int* __restrict__ params) {
  if (threadIdx.x != 0) return;
  const int b = blockIdx.x;
  const unsigned int* h = hist + (b << 16);
  unsigned int cum = 0;
  int bin = 65535;
  for (; bin > 0; --bin) {
    const unsigned int c = h[bin];
    if (cum + c >= (unsigned)KSEL) break;
    cum += c;
  }
  params[b * 8 + 0] = (unsigned)bin;
  params[b * 8 + 1] = cum;
  params[b * 8 + 2] = (unsigned)KSEL - cum;
}

// ---------------- pass 2: histogram of low 16 bits within threshold bin ----
__global__ void hist_lo_kernel(const float* __restrict__ probs,
                               const unsigned int* __restrict__ params,
                               unsigned int* __restrict__ hist2) {
  const int n = BATCH * NA;
  for (int i = blockIdx.x * blockDim.x + threadIdx.x; i < n;
       i += gridDim.x * blockDim.x) {
    const int b = i / NA;
    const unsigned int bits = __float_as_uint(probs[2 * i + 1]);
    if ((bits >> 16) == params[b * 8 + 0])
      atomicAdd(&hist2[(b << 16) + (bits & 0xFFFFu)], 1u);
  }
}

__global__ void find_lo_kernel(const unsigned int* __restrict__ hist2,
                               unsigned int* __restrict__ params) {
  if (threadIdx.x != 0) return;
  const int b = blockIdx.x;
  const unsigned int* h = hist2 + (b << 16);
  const unsigned int need = params[b * 8 + 2];
  unsigned int cum = 0;
  int bin = 65535;
  for (; bin > 0; --bin) {
    const unsigned int c = h[bin];
    if (cum + c >= need) break;
    cum += c;
  }
  const unsigned int gt = params[b * 8 + 1] + cum;     // strictly > vstar
  params[b * 8 + 3] = (params[b * 8 + 0] << 16) | (unsigned)bin;
  params[b * 8 + 4] = gt;
  params[b * 8 + 5] = (unsigned)KSEL - gt;             // equals to take
}

// ---------------- compaction: >vstar into candidates, ==vstar into tie list -
__global__ void compact_kernel(const float* __restrict__ probs,
                               unsigned int* __restrict__ params,
                               unsigned int* __restrict__ candIdx,
                               float* __restrict__ candScore,
                               unsigned int* __restrict__ equals) {
  const int n = BATCH * NA;
  for (int i = blockIdx.x * blockDim.x + threadIdx.x; i < n;
       i += gridDim.x * blockDim.x) {
    const int b = i / NA;
    const int a = i - b * NA;
    const float s = probs[2 * i + 1];
    const unsigned int bits = __float_as_uint(s);
    const unsigned int vstar = params[b * 8 + 3];
    if (bits > vstar) {
      const unsigned int slot = atomicAdd(&params[b * 8 + 6], 1u);
      if (slot < (unsigned)KSEL) {
        candIdx[b * KSEL + slot]   = (unsigned)a;
        candScore[b * KSEL + slot] = s;
      }
    } else if (bits == vstar) {
      const unsigned int e = atomicAdd(&params[b * 8 + 7], 1u);
      if (e < (unsigned)ECAP) equals[b * ECAP + e] = (unsigned)a;
    }
  }
}

// tie-break exactly like jax top_k: among equal scores, lower index first.
__global__ void resolve_equals_kernel(const unsigned int* __restrict__ equals,
                                      const unsigned int* __restrict__ params,
                                      unsigned int* __restrict__ candIdx,
                                      float* __restrict__ candScore) {
  __shared__ unsigned int se[ECAP];
  const int b = blockIdx.x;
  const int t = threadIdx.x;
  unsigned int E = params[b * 8 + 7];
  if (E > (unsigned)ECAP) E = ECAP;
  const unsigned int R    = params[b * 8 + 5];
  const unsigned int base = params[b * 8 + 4];
  const float vs = __uint_as_float(params[b * 8 + 3]);
  for (unsigned int i = t; i < E; i += blockDim.x) se[i] = equals[b * ECAP + i];
  __syncthreads();
  for (unsigned int i = t; i < E; i += blockDim.x) {
    const unsigned int mine = se[i];
    unsigned int rank = 0;
    for (unsigned int j = 0; j < E; ++j) rank += (se[j] < mine) ? 1u : 0u;
    if (rank < R && base + rank < (unsigned)KSEL) {
      candIdx[b * KSEL + base + rank]   = mine;
      candScore[b * KSEL + base + rank] = vs;
    }
  }
}

// ---------------- rank-sort candidates by (score desc, idx asc) in LDS ------
// LDS fill uses gfx1250 async global->LDS DMA (GLOBAL_LOAD_ASYNC_TO_LDS_B128,
// tracked by ASYNCcnt): SGPR base + per-lane 32-bit byte offset (GVS mode),
// LDS byte address carried in the VDST VGPR. No VGPR round-trip for the fill.
__global__ void rank_sort_kernel(const unsigned int* __restrict__ candIdx,
                                 const float* __restrict__ candScore,
                                 unsigned int* __restrict__ sortedIdx) {
  __shared__ __align__(16) unsigned int sb[KSEL];  // score bits
  __shared__ __align__(16) unsigned int si[KSEL];  // anchor indices
  const int b = blockIdx.x, slice = blockIdx.y;
  const int t = threadIdx.x;

  const float* gs        = candScore + (size_t)b * KSEL;
  const unsigned int* gi = candIdx   + (size_t)b * KSEL;
  const unsigned int sbase = (unsigned int)(size_t)(void*)sb;
  const unsigned int ibase = (unsigned int)(size_t)(void*)si;
  for (int c = t * 4; c < KSEL; c += (int)blockDim.x * 4) {
    const unsigned int lds_s = sbase + (unsigned)c * 4u;
    const unsigned int lds_i = ibase + (unsigned)c * 4u;
    const int goff = c * 4;                       // byte offset into batch row
    asm volatile("global_load_async_to_lds_b128 %0, %1, %2"
                 :: "v"(lds_s), "v"(goff), "s"(gs) : "memory");
    asm volatile("global_load_async_to_lds_b128 %0, %1, %2"
                 :: "v"(lds_i), "v"(goff), "s"(gi) : "memory");
  }
  asm volatile("s_wait_asynccnt 0" ::: "memory");
  __syncthreads();

  const int span = KSEL / 8;
  const int c0 = slice * span, c1 = c0 + span;
  for (int c = c0 + t; c < c1; c += blockDim.x) {
    const unsigned int mys = sb[c], myi = si[c];
    int rank = 0;
    for (int j = 0; j < KSEL; ++j) {
      const unsigned int s2 = sb[j];
      rank += ((s2 > mys) | ((s2 == mys) & (si[j] < myi))) ? 1 : 0;
    }
    sortedIdx[b * KSEL + rank] = myi;
  }
}

// ---------------- gather + apply box deltas ----------------
__global__ void gather_boxes_kernel(const unsigned int* __restrict__ sortedIdx,
                                    const float* __restrict__ rpn_bbox,
                                    const float* __restrict__ anchors,
                                    float* __restrict__ boxes) {
  const int n = BATCH * KSEL;
  for (int i = blockIdx.x * blockDim.x + threadIdx.x; i < n;
       i += gridDim.x * blockDim.x) {
    const int b = i / KSEL;
    unsigned int a = sortedIdx[i];
    if (a >= (unsigned)NA) a = 0;               // safety clamp
    const size_t base = ((size_t)b * NA + a) * 4;
    const float d0 = rpn_bbox[base + 0] * 0.1f;
    const float d1 = rpn_bbox[base + 1] * 0.1f;
    const float d2 = rpn_bbox[base + 2] * 0.2f;
    const float d3 = rpn_bbox[base + 3] * 0.2f;
    const float y1 = anchors[base + 0], x1 = anchors[base + 1];
    const float y2 = anchors[base + 2], x2 = anchors[base + 3];
    float h = y2 - y1, w = x2 - x1;
    const float cy = y1 + 0.5f * h + d0 * h;
    const float cx = x1 + 0.5f * w + d1 * w;
    h = h * expf(d2);
    w = w * expf(d3);
    const float ny1 = cy - 0.5f * h;
    const float nx1 = cx - 0.5f * w;
    float4 o = make_float4(ny1, nx1, ny1 + h, nx1 + w);
    *(float4*)(boxes + (size_t)i * 4) = o;
  }
}

// ---------------- suppression bitmask: one wave = 16 rows x 32 cols ---------
// Union term (area_i + area_j) of each 16x16 tile is computed on the matrix
// pipe via V_WMMA_F32_16X16X4_F32 (A=[area_row,1], B=[1;area_col]); the VALU
// does intersection min/max; wave32 ballot packs 32 columns into one u32.
// Predicates are fully branchless: column validity is applied as a ballot
// mask AND, so the hot loop has zero EXEC manipulation.
__global__ void nms_mask_kernel(const float* __restrict__ boxesAll,
                                unsigned int* __restrict__ maskAll) {
  const int lane = threadIdx.x;
  const int r0 = blockIdx.x * 16;
  const int tc = blockIdx.y;
  const int b  = blockIdx.z;
  const float* boxes = boxesAll + (size_t)b * KSEL * 4;
  unsigned int* mask = maskAll + (size_t)b * KSEL * NW;

  __shared__ float rb0[16], rb1[16], rb2[16], rb3[16], rarea[16];
  if (lane < 16) {
    const float* p = boxes + (size_t)(r0 + lane) * 4;
    const float y1 = p[0], x1 = p[1], y2 = p[2], x2 = p[3];
    rb0[lane] = y1; rb1[lane] = x1; rb2[lane] = y2; rb3[lane] = x2;
    rarea[lane] = (y2 - y1) * (x2 - x1);
  }
  __syncthreads();

  const int cbase = tc * 32;
  const int csub  = lane & 15;
  const int col0  = cbase + csub;
  const int col1  = cbase + 16 + csub;
  const bool valid0 = col0 < KSEL;
  const bool valid1 = col1 < KSEL;
  // column-validity as wave-uniform bit masks (applied after ballot)
  const unsigned int vm0 = __builtin_amdgcn_ballot_w32(valid0);
  const unsigned int vm1 = __builtin_amdgcn_ballot_w32(valid1);
  const float* q0 = boxes + (size_t)(valid0 ? col0 : 0) * 4;
  const float* q1 = boxes + (size_t)(valid1 ? col1 : 0) * 4;
  const float a0y1 = q0[0], a0x1 = q0[1], a0y2 = q0[2], a0x2 = q0[3];
  const float a1y1 = q1[0], a1x1 = q1[1], a1y2 = q1[2], a1x2 = q1[3];
  const float carea0 = (a0y2 - a0y1) * (a0x2 - a0x1);
  const float carea1 = (a1y2 - a1y1) * (a1x2 - a1x1);

  const bool lo = lane < 16;
  // A (16x4): K0 = area_row, K1 = 1, K2 = K3 = 0  (lanes 16-31 hold K2/K3)
  v2f A;  A.x  = lo ? rarea[lane & 15] : 0.f;  A.y  = lo ? 1.f : 0.f;
  // B (4x16): K0 = 1, K1 = area_col, K2 = K3 = 0
  v2f B0; B0.x = lo ? 1.f : 0.f;               B0.y = lo ? carea0 : 0.f;
  v2f B1; B1.x = lo ? 1.f : 0.f;               B1.y = lo ? carea1 : 0.f;
  v8f Cz = {0.f, 0.f, 0.f, 0.f, 0.f, 0.f, 0.f, 0.f};

  // D[M][N] = area_row[M] + area_col[N]  (exact: two nonzero products)
  v8f S0 = __builtin_amdgcn_wmma_f32_16x16x4_f32(false, A, false, B0,
                                                 (short)0, Cz, false, false);
  v8f S1 = __builtin_amdgcn_wmma_f32_16x16x4_f32(false, A, false, B1,
                                                 (short)0, Cz, false, false);

  unsigned int bal0[8], bal1[8];
#pragma unroll
  for (int r = 0; r < 8; ++r) {
    const int M = lo ? r : (8 + r);     // C/D layout: lanes 0-15 -> M=r, 16-31 -> M=8+r
    const float ry1 = rb0[M], rx1 = rb1[M], ry2 = rb2[M], rx2 = rb3[M];

    float iy1 = fmaxf(ry1, a0y1), ix1 = fmaxf(rx1, a0x1);
    float iy2 = fminf(ry2, a0y2), ix2 = fminf(rx2, a0x2);
    float inter = fmaxf(iy2 - iy1, 0.f) * fmaxf(ix2 - ix1, 0.f);
    float uni = S0[r] - inter;
    bal0[r] = __builtin_amdgcn_ballot_w32(inter > NMS_T * fmaxf(uni, 1e-12f)) & vm0;

    iy1 = fmaxf(ry1, a1y1); ix1 = fmaxf(rx1, a1x1);
    iy2 = fminf(ry2, a1y2); ix2 = fminf(rx2, a1x2);
    inter = fmaxf(iy2 - iy1, 0.f) * fmaxf(ix2 - ix1, 0.f);
    uni = S1[r] - inter;
    bal1[r] = __builtin_amdgcn_ballot_w32(inter > NMS_T * fmaxf(uni, 1e-12f)) & vm1;
  }

  if (lane == 0) {
#pragma unroll
    for (int r = 0; r < 8; ++r) {
      // ballot bits 0-15: rows M=r (cols cbase+l); bits 16-31: rows M=8+r
      const unsigned int wlow  = (bal0[r] & 0xFFFFu) | ((bal1[r] & 0xFFFFu) << 16);
      const unsigned int whigh = (bal0[r] >> 16)     | (bal1[r] & 0xFFFF0000u);
      mask[(size_t)(r0 + r)     * NW + tc] = wlow;
      mask[(size_t)(r0 + 8 + r) * NW + tc] = whigh;
    }
  }
}

// ---------------- sequential greedy sweep + compaction, one block per batch -
__global__ void nms_sweep_kernel(const unsigned int* __restrict__ maskAll,
                                 const float* __restrict__ boxesAll,
                                 float* __restrict__ outAll) {
  const int b = blockIdx.x;
  const unsigned int* mask = maskAll + (size_t)b * KSEL * NW;
  const float* boxes = boxesAll + (size_t)b * KSEL * 4;
  float* out = outAll + (size_t)b * NOUT * 4;

  __shared__ unsigned int keep[NW];
  __shared__ unsigned int pfx[NW];
  const int t = threadIdx.x;
  for (int w = t; w < NW; w += blockDim.x)
    keep[w] = (w == NW - 1) ? 0x0000FFFFu : 0xFFFFFFFFu;  // mask tail cols >= KSEL
  __syncthreads();

  for (int i = 0; i < KSEL; ++i) {
    const int wi = i >> 5, bi = i & 31;
    const unsigned int ki = keep[wi];   // bit bi is invariant this iteration
    __syncthreads();
    if ((ki >> bi) & 1u) {
      const unsigned int* mrow = mask + (size_t)i * NW;
      for (int w = t; w < NW; w += blockDim.x) {
        unsigned int m = mrow[w];
        if (w < wi) m = 0u;
        else if (w == wi) m &= (bi < 31) ? (0xFFFFFFFFu << (bi + 1)) : 0u; // j > i only
        keep[w] &= ~m;
      }
    }
    __syncthreads();
  }

  if (t == 0) {
    unsigned int s = 0;
    for (int w = 0; w < NW; ++w) { pfx[w] = s; s += __popc(keep[w]); }
  }
  __syncthreads();

  for (int w = t; w < NW; w += blockDim.x) {
    unsigned int k = keep[w];
    unsigned int pos = pfx[w];
    while (k) {
      const int bit = __ffs(k) - 1;
      k &= k - 1;
      if (pos < (unsigned)NOUT) {
        const float4 v = *(const float4*)(boxes + (size_t)(w * 32 + bit) * 4);
        *(float4*)(out + (size_t)pos * 4) = v;
      }
      ++pos;
    }
  }
}

// ---------------- host-side orchestration ----------------
extern "C" void kernel_launch(void* const* d_in, const int* in_sizes, int n_in,
                              void* d_out, int out_size, void* d_ws, size_t ws_size,
                              hipStream_t stream) {
  (void)in_sizes; (void)n_in; (void)out_size; (void)ws_size;
  const float* rpn_probs = (const float*)d_in[0];   // (8, 261888, 2)
  const float* rpn_bbox  = (const float*)d_in[1];   // (8, 261888, 4)
  const float* anchors   = (const float*)d_in[2];   // (8, 261888, 4)
  float* out = (float*)d_out;                       // (8, 2000, 4)

  // bump-allocate workspace (256B aligned)
  char* ws = (char*)d_ws;
  size_t off = 0;
  auto alloc = [&](size_t bytes) -> char* {
    char* p = ws + off;
    off = (off + bytes + 255) & ~(size_t)255;
    return p;
  };
  unsigned int* hist1     = (unsigned int*)alloc((size_t)BATCH * 65536 * 4); // 2 MB
  unsigned int* hist2     = (unsigned int*)alloc((size_t)BATCH * 65536 * 4); // 2 MB
  unsigned int* params    = (unsigned int*)alloc((size_t)BATCH * 8 * 4);
  unsigned int* candIdx   = (unsigned int*)alloc((size_t)BATCH * KSEL * 4);
  float*        candScore = (float*)       alloc((size_t)BATCH * KSEL * 4);
  unsigned int* equals    = (unsigned int*)alloc((size_t)BATCH * ECAP * 4);
  unsigned int* sortedIdx = (unsigned int*)alloc((size_t)BATCH * KSEL * 4);
  float*        boxes     = (float*)       alloc((size_t)BATCH * KSEL * 4 * 4);
  unsigned int* mask      = (unsigned int*)alloc((size_t)BATCH * KSEL * NW * 4); // 36 MB

  // zero hist1..params span (contiguous in the bump allocator) and the output
  {
    const int zn = (int)(((char*)params + BATCH * 8 * 4 - (char*)hist1) / 4);
    zero_u32_kernel<<<2048, 256, 0, stream>>>(hist1, zn);
    zero_u32_kernel<<<64, 256, 0, stream>>>((unsigned int*)out, BATCH * NOUT * 4);
  }

  // exact top-6000 via two-level 16-bit radix select
  hist_hi_kernel<<<4096, 256, 0, stream>>>(rpn_probs, hist1);
  find_hi_kernel<<<BATCH, 32, 0, stream>>>(hist1, params);
  hist_lo_kernel<<<4096, 256, 0, stream>>>(rpn_probs, params, hist2);
  find_lo_kernel<<<BATCH, 32, 0, stream>>>(hist2, params);
  compact_kernel<<<4096, 256, 0, stream>>>(rpn_probs, params, candIdx,
                                           candScore, equals);
  resolve_equals_kernel<<<BATCH, 256, 0, stream>>>(equals, params, candIdx,
                                                   candScore);

  // deterministic sort of the 6000 candidates: rank by (score desc, idx asc)
  rank_sort_kernel<<<dim3(BATCH, 8), 512, 0, stream>>>(candIdx, candScore,
                                                       sortedIdx);

  // gather anchors/deltas and apply box regression
  gather_boxes_kernel<<<192, 256, 0, stream>>>(sortedIdx, rpn_bbox, anchors,
                                               boxes);

  // pairwise suppression bitmask (WMMA union-sum + wave32 ballots), then the
  // serial greedy sweep — all 8 batches concurrently.
  nms_mask_kernel<<<dim3(KSEL / 16, NW, BATCH), 32, 0, stream>>>(boxes, mask);
  nms_sweep_kernel<<<BATCH, 256, 0, stream>>>(mask, boxes, out);
}